// NeuralImageFunction_68728066670741
// MI455X (gfx1250) — compile-verified
//
#include <hip/hip_runtime.h>
#include <hip/hip_bf16.h>

typedef __attribute__((ext_vector_type(2))) float v2f;
typedef __attribute__((ext_vector_type(8))) float v8f;

#define H_ 1024
#define W_ 1024
#define RNK 128
#define MAXRANK 204
#define NROW 384            // 3 channels * 128 ranks

__device__ __forceinline__ v8f wmma_f32(v2f a, v2f b, v8f c) {
    // V_WMMA_F32_16X16X4_F32 : D = A(16x4,f32) * B(4x16,f32) + C(16x16,f32)
    return __builtin_amdgcn_wmma_f32_16x16x4_f32(
        false, a, false, b, (short)0, c, false, false);
}

// ---------------------------------------------------------------------------
// K0: build DFT matrix F[n,k] = exp(-2*pi*i*n*k/1024) and phase matrix
//     P[r,x] = exp(+2*i*PIref*r*x/1024), PIref = 3.141592 (reference constant)
// ---------------------------------------------------------------------------
__global__ void build_tables(float* __restrict__ Fre, float* __restrict__ Fim,
                             float* __restrict__ Pre, float* __restrict__ Pim) {
    long i = (long)blockIdx.x * blockDim.x + threadIdx.x;
    const long nF = (long)H_ * W_;
    if (i < nF) {
        int n = (int)(i >> 10), k = (int)(i & 1023);
        // exact mod-1024 reduction is mathematically exact for the true DFT
        float ang = -6.2831853071795864f * (float)((n * k) & 1023) * (1.0f / 1024.0f);
        Fre[i] = cosf(ang);
        Fim[i] = sinf(ang);
    } else if (i < nF + (long)RNK * H_) {
        long j = i - nF;
        int r = (int)(j >> 10), x = (int)(j & 1023);
        float ang = 6.283184f * ((float)(r * x) * (1.0f / 1024.0f)); // 2*3.141592
        Pre[j] = cosf(ang);
        Pim[j] = sinf(ang);
    }
}

// ---------------------------------------------------------------------------
// K1: Z = X * F  (complex GEMM, M=384 rows per batch, K=1024, N=1024)
//     Two batches (z1 from rank1_*, z2 from rank2_*). One wave per 16x16 tile.
// ---------------------------------------------------------------------------
__global__ void dft_gemm(const float* __restrict__ r1re, const float* __restrict__ r1im,
                         const float* __restrict__ r2re, const float* __restrict__ r2im,
                         const float* __restrict__ Fre, const float* __restrict__ Fim,
                         float* __restrict__ Zre, float* __restrict__ Zim) {
    const int lane = threadIdx.x & 31;
    const int wid  = (int)((blockIdx.x * blockDim.x + threadIdx.x) >> 5);
    // 2 batches * 24 mtiles * 64 ntiles = 3072 waves
    const int zi  = wid / (24 * 64);
    const int rem = wid % (24 * 64);
    const int mt  = rem >> 6;
    const int nt  = rem & 63;

    const float* Xre = zi ? r2re : r1re;
    const float* Xim = zi ? r2im : r1im;

    const int m   = lane & 15;
    const int row = mt * 16 + m;                 // 0..383 -> (c, r)
    const int c   = row >> 7;
    const int r   = row & 127;
    const long inOff = ((long)(c * MAXRANK + r)) << 10;   // row stride 1024, rank stride 204
    const int koff = (lane >> 4) << 1;           // lanes 16-31 hold K+2,K+3
    const int n    = nt * 16 + (lane & 15);

    v8f accRR = {}, accII = {}, accRI = {}, accIR = {};
    #pragma unroll 2
    for (int kk = 0; kk < 1024; kk += 4) {
        const int ka = kk + koff;
        v2f are = *(const v2f*)(Xre + inOff + ka);
        v2f aim = *(const v2f*)(Xim + inOff + ka);
        v2f bre, bim;
        bre.x = Fre[ka * W_ + n];       bre.y = Fre[(ka + 1) * W_ + n];
        bim.x = Fim[ka * W_ + n];       bim.y = Fim[(ka + 1) * W_ + n];
        accRR = wmma_f32(are, bre, accRR);
        accII = wmma_f32(aim, bim, accII);
        accRI = wmma_f32(are, bim, accRI);
        accIR = wmma_f32(aim, bre, accIR);
    }
    // C/D layout: VGPR i -> M = i (lanes 0-15) / i+8 (lanes 16-31), N = lane&15
    const int col      = nt * 16 + (lane & 15);
    const int rowHalf  = (lane >> 4) * 8;
    const long zbase   = (long)zi * NROW * W_;
    #pragma unroll
    for (int i = 0; i < 8; ++i) {
        const int orow = mt * 16 + rowHalf + i;
        const long o = zbase + (long)orow * W_ + col;
        Zre[o] = accRR[i] - accII[i];
        Zim[o] = accRI[i] + accIR[i];
    }
}

// ---------------------------------------------------------------------------
// K2: 9-tap Gaussian correlation (zero SAME padding) + complex recombine
//     Cre = ca - cb, Cim = ca + cb
// ---------------------------------------------------------------------------
__global__ void conv9(const float* __restrict__ Zre, const float* __restrict__ Zim,
                      const float* __restrict__ sigma,
                      float* __restrict__ Cre, float* __restrict__ Cim) {
    const int idx = blockIdx.x * blockDim.x + threadIdx.x;   // 768*1024 threads
    if (idx >= 2 * NROW * W_) return;
    const int rowBase = idx & ~1023;
    const int col     = idx & 1023;
    const float sg    = sigma[0];
    const float s     = fmaxf(sg, 0.1f);
    const float inv   = 1.0f / fmaxf(sg * 2.5066282f, 1.0f);  // sqrt(2*pi) f32
    float ca = 0.0f, cb = 0.0f;
    #pragma unroll
    for (int j = 0; j < 9; ++j) {
        const float t  = (float)(j - 4) / s;
        const float kw = expf(-0.5f * t * t) * inv;
        const int cc = col + j - 4;
        if (cc >= 0 && cc < 1024) {
            ca += kw * Zre[rowBase + cc];
            cb += kw * Zim[rowBase + cc];
        }
    }
    Cre[idx] = ca - cb;
    Cim[idx] = ca + cb;
}

// ---------------------------------------------------------------------------
// K3: rgb[c,h,w] = 255*( sum_r Pre[r,h]*C1re[c,r,w] - Pim[r,h]*C1im[c,r,w]
//                      + sum_r C2re[c,r,h]*Pre[r,w] - C2im[c,r,h]*Pim[r,w] )
//     K=128 GEMMs, one wave per 16x16 tile, 4 WMMA chains.
// ---------------------------------------------------------------------------
__global__ void phase_gemm(const float* __restrict__ Cre, const float* __restrict__ Cim,
                           const float* __restrict__ Pre, const float* __restrict__ Pim,
                           float* __restrict__ rgb) {
    const int lane = threadIdx.x & 31;
    const int wid  = (int)((blockIdx.x * blockDim.x + threadIdx.x) >> 5);
    // 3 channels * 64 * 64 = 12288 waves
    const int cch = wid >> 12;
    const int rem = wid & 4095;
    const int mt  = rem >> 6;
    const int nt  = rem & 63;

    const int koff = (lane >> 4) << 1;
    const int hm = mt * 16 + (lane & 15);   // M (h) index for A fragments
    const int wn = nt * 16 + (lane & 15);   // N (w) index for B fragments

    const float* C1re = Cre;
    const float* C1im = Cim;
    const float* C2re = Cre + (long)NROW * W_;
    const float* C2im = Cim + (long)NROW * W_;
    const long cbase = (long)cch * RNK * W_;

    v8f a1 = {}, a2 = {}, a3 = {}, a4 = {};
    #pragma unroll 4
    for (int kk = 0; kk < RNK; kk += 4) {
        const int kr = kk + koff;
        v2f aP, aQ, aR, aS, b1, b2, b3, b4;
        aP.x = Pre[kr * W_ + hm];             aP.y = Pre[(kr + 1) * W_ + hm];
        aQ.x = Pim[kr * W_ + hm];             aQ.y = Pim[(kr + 1) * W_ + hm];
        aR.x = C2re[cbase + kr * W_ + hm];    aR.y = C2re[cbase + (kr + 1) * W_ + hm];
        aS.x = C2im[cbase + kr * W_ + hm];    aS.y = C2im[cbase + (kr + 1) * W_ + hm];
        b1.x = C1re[cbase + kr * W_ + wn];    b1.y = C1re[cbase + (kr + 1) * W_ + wn];
        b2.x = C1im[cbase + kr * W_ + wn];    b2.y = C1im[cbase + (kr + 1) * W_ + wn];
        b3.x = Pre[kr * W_ + wn];             b3.y = Pre[(kr + 1) * W_ + wn];
        b4.x = Pim[kr * W_ + wn];             b4.y = Pim[(kr + 1) * W_ + wn];
        a1 = wmma_f32(aP, b1, a1);
        a2 = wmma_f32(aQ, b2, a2);
        a3 = wmma_f32(aR, b3, a3);
        a4 = wmma_f32(aS, b4, a4);
    }
    const int col     = nt * 16 + (lane & 15);
    const int rowHalf = (lane >> 4) * 8;
    const long obase  = (long)cch * H_ * W_;
    #pragma unroll
    for (int i = 0; i < 8; ++i) {
        const int orow = mt * 16 + rowHalf + i;
        rgb[obase + (long)orow * W_ + col] =
            255.0f * (a1[i] - a2[i] + a3[i] - a4[i]);
    }
}

// ---------------------------------------------------------------------------
// K4: bilinear grid sample with zero-outside boundary; out shape (B, N, 3)
// ---------------------------------------------------------------------------
__global__ void bilerp(const float* __restrict__ coord, const float* __restrict__ rgb,
                       float* __restrict__ out, long total) {
    const long i = (long)blockIdx.x * blockDim.x + threadIdx.x;
    if (i >= total) return;
    const float gx = coord[2 * i + 0];
    const float gy = coord[2 * i + 1];
    const float ix = ((gx + 1.0f) * (float)W_ - 1.0f) * 0.5f;
    const float iy = ((gy + 1.0f) * (float)H_ - 1.0f) * 0.5f;
    const float ix0 = floorf(ix), iy0 = floorf(iy);
    const float ix1 = ix0 + 1.0f, iy1 = iy0 + 1.0f;
    const float wx1 = ix - ix0, wx0 = 1.0f - wx1;
    const float wy1 = iy - iy0, wy0 = 1.0f - wy1;

    const float xs[2] = {ix0, ix1};
    const float ys[2] = {iy0, iy1};
    const float wx[2] = {wx0, wx1};
    const float wy[2] = {wy0, wy1};

    float acc0 = 0.0f, acc1 = 0.0f, acc2 = 0.0f;
    #pragma unroll
    for (int ty = 0; ty < 2; ++ty) {
        #pragma unroll
        for (int tx = 0; tx < 2; ++tx) {
            const float xf = xs[tx], yf = ys[ty];
            const float w  = wx[tx] * wy[ty];
            const bool valid = (xf >= 0.0f) && (xf <= (float)(W_ - 1)) &&
                               (yf >= 0.0f) && (yf <= (float)(H_ - 1));
            const float wv = valid ? w : 0.0f;
            int xc = (int)xf; xc = xc < 0 ? 0 : (xc > W_ - 1 ? W_ - 1 : xc);
            int yc = (int)yf; yc = yc < 0 ? 0 : (yc > H_ - 1 ? H_ - 1 : yc);
            const long p = (long)yc * W_ + xc;
            acc0 += wv * rgb[p];
            acc1 += wv * rgb[(long)H_ * W_ + p];
            acc2 += wv * rgb[2L * H_ * W_ + p];
        }
    }
    out[3 * i + 0] = acc0;
    out[3 * i + 1] = acc1;
    out[3 * i + 2] = acc2;
}

// ---------------------------------------------------------------------------
extern "C" void kernel_launch(void* const* d_in, const int* in_sizes, int n_in,
                              void* d_out, int out_size, void* d_ws, size_t ws_size,
                              hipStream_t stream) {
    const float* r1re  = (const float*)d_in[0];
    const float* r1im  = (const float*)d_in[1];
    const float* r2re  = (const float*)d_in[2];
    const float* r2im  = (const float*)d_in[3];
    const float* sigma = (const float*)d_in[4];
    const float* coord = (const float*)d_in[5];
    (void)in_sizes; (void)n_in; (void)out_size; (void)ws_size;

    float* ws = (float*)d_ws;
    // workspace layout (floats)
    float* Fre = ws;                                 // 1024*1024
    float* Fim = Fre + (long)H_ * W_;                // 1024*1024
    float* Pre = Fim + (long)H_ * W_;                // 128*1024
    float* Pim = Pre + (long)RNK * W_;               // 128*1024
    float* Zre = Pim + (long)RNK * W_;               // 2*384*1024
    float* Zim = Zre + 2L * NROW * W_;               // 2*384*1024
    float* Cre = Zim + 2L * NROW * W_;               // 2*384*1024
    float* Cim = Cre + 2L * NROW * W_;               // 2*384*1024
    float* rgb = Cim + 2L * NROW * W_;               // 3*1024*1024

    // K0: twiddle / phase tables (1M + 128K threads)
    {
        const long n = (long)H_ * W_ + (long)RNK * H_;
        build_tables<<<(int)((n + 255) / 256), 256, 0, stream>>>(Fre, Fim, Pre, Pim);
    }
    // K1: DFT as fp32-WMMA complex GEMM (3072 waves, 8 waves/block)
    dft_gemm<<<384, 256, 0, stream>>>(r1re, r1im, r2re, r2im, Fre, Fim, Zre, Zim);
    // K2: Gaussian 9-tap conv + recombine
    conv9<<<(2 * NROW * W_) / 256, 256, 0, stream>>>(Zre, Zim, sigma, Cre, Cim);
    // K3: phase einsums as fp32-WMMA GEMM, fused *255 epilogue (12288 waves)
    phase_gemm<<<1536, 256, 0, stream>>>(Cre, Cim, Pre, Pim, rgb);
    // K4: bilinear sample, output (B,N,3)
    {
        const long total = 8L * 262144L;
        bilerp<<<(int)((total + 255) / 256), 256, 0, stream>>>(coord, rgb, (float*)d_out, total);
    }
}